// MotoneuronNetwork_66941360275495
// MI455X (gfx1250) — compile-verified
//
#include <hip/hip_runtime.h>
#include <hip/hip_bf16.h>
#include <stdint.h>

typedef float v2f __attribute__((ext_vector_type(2)));
typedef float v8f __attribute__((ext_vector_type(8)));

// Problem geometry (Pinsky-Rinzel network: S samples x 1024 neurons x 8 states)
static constexpr int NEUR = 1024;   // K and N of the GEMM
static constexpr int BM   = 128;    // sample tile
static constexpr int BN   = 64;     // neuron tile
static constexpr int BK   = 16;     // K tile
static constexpr int LDA  = 20;     // padded LDS row stride (floats), conflict-free b64 reads
static constexpr int LDB  = 20;

// CDNA5 async global->LDS copy (no VGPR roundtrip, tracked by ASYNCcnt).
__device__ __forceinline__ void lds_async_b128(unsigned lds_off, const void* g) {
    asm volatile("global_load_async_to_lds_b128 %0, %1, off"
                 :: "v"(lds_off), "v"((unsigned long long)(uintptr_t)g)
                 : "memory");
}
__device__ __forceinline__ void wait_asynccnt0() {
    asm volatile("s_wait_asynccnt 0x0" ::: "memory");
}

__global__ __launch_bounds__(256)
void pr_fused_wmma_kernel(const float* __restrict__ y,
                          const float* __restrict__ I_ext,
                          const float* __restrict__ synI,
                          const float* __restrict__ w,
                          const int*   __restrict__ net,
                          float* __restrict__ out)
{
    __shared__ float Asm[2][BM * LDA];   // A tile: synaptic_I [m][k], double buffered
    __shared__ float Bsm[2][BN * LDB];   // B tile: (w*net)^T as [n][k], double buffered

    const int tid   = threadIdx.x;
    const int lane  = tid & 31;
    const int wid   = tid >> 5;        // 8 waves
    const int lm    = lane & 15;
    const int hi    = lane >> 4;       // half-wave select (K pairs / M+8 rows)
    const int waveM = wid & 3;         // 4 waves along M  -> 32 rows each
    const int waveN = wid >> 2;        // 2 waves along N  -> 32 cols each
    const int mbase = blockIdx.x * BM;
    const int nbase = blockIdx.y * BN;

    // Per-thread staging coordinates (one float4 lane-slot each)
    const int st_row = tid >> 2;          // 0..63
    const int st_k   = (tid & 3) << 2;    // 0,4,8,12

    v8f acc[2][2] = {{{}, {}}, {{}, {}}};   // 2x2 subtiles of 16x16
    const int am0 = waveM * 32;
    const int bn0 = waveN * 32;

    // ---- prologue: stage buffer 0 ----
    lds_async_b128((unsigned)(uintptr_t)&Asm[0][st_row * LDA + st_k],
                   synI + (size_t)(mbase + st_row) * NEUR + st_k);
    lds_async_b128((unsigned)(uintptr_t)&Asm[0][(st_row + 64) * LDA + st_k],
                   synI + (size_t)(mbase + st_row + 64) * NEUR + st_k);
    {
        const size_t g = (size_t)(nbase + st_row) * NEUR + st_k;
        const float4 wv = *reinterpret_cast<const float4*>(w + g);
        const int4   mv = *reinterpret_cast<const int4*>(net + g);
        float4 bv;
        bv.x = wv.x * (float)mv.x;
        bv.y = wv.y * (float)mv.y;
        bv.z = wv.z * (float)mv.z;
        bv.w = wv.w * (float)mv.w;
        *reinterpret_cast<float4*>(&Bsm[0][st_row * LDB + st_k]) = bv;
    }
    wait_asynccnt0();
    __syncthreads();

    // ---- software-pipelined K loop ----
    for (int k0 = 0; k0 < NEUR; k0 += BK) {
        const int  cur  = (k0 >> 4) & 1;
        const int  nxt  = cur ^ 1;
        const bool more = (k0 + BK) < NEUR;

        float4 wv; int4 mv;
        if (more) {
            const int k1 = k0 + BK;
            // async A copy for next tile runs under this tile's WMMAs
            lds_async_b128((unsigned)(uintptr_t)&Asm[nxt][st_row * LDA + st_k],
                           synI + (size_t)(mbase + st_row) * NEUR + k1 + st_k);
            lds_async_b128((unsigned)(uintptr_t)&Asm[nxt][(st_row + 64) * LDA + st_k],
                           synI + (size_t)(mbase + st_row + 64) * NEUR + k1 + st_k);
            // B loads issued now, consumed (mask+store) after the WMMA block
            const size_t g = (size_t)(nbase + st_row) * NEUR + k1 + st_k;
            wv = *reinterpret_cast<const float4*>(w + g);
            mv = *reinterpret_cast<const int4*>(net + g);
        }

        // ---- 16 x V_WMMA_F32_16X16X4_F32 on current buffer ----
        #pragma unroll
        for (int k = 0; k < BK; k += 4) {
            // A frag (16x4): lanes0-15 -> M=lane,K={k,k+1}; lanes16-31 -> K={k+2,k+3}
            const v2f a0 = *reinterpret_cast<const v2f*>(&Asm[cur][(am0 + lm     ) * LDA + k + 2 * hi]);
            const v2f a1 = *reinterpret_cast<const v2f*>(&Asm[cur][(am0 + 16 + lm) * LDA + k + 2 * hi]);
            // B frag (4x16): lanes0-15 -> N=lane,K={k,k+1}; lanes16-31 -> K={k+2,k+3}
            const v2f b0 = *reinterpret_cast<const v2f*>(&Bsm[cur][(bn0 + lm     ) * LDB + k + 2 * hi]);
            const v2f b1 = *reinterpret_cast<const v2f*>(&Bsm[cur][(bn0 + 16 + lm) * LDB + k + 2 * hi]);
            acc[0][0] = __builtin_amdgcn_wmma_f32_16x16x4_f32(false, a0, false, b0, (short)0, acc[0][0], false, false);
            acc[0][1] = __builtin_amdgcn_wmma_f32_16x16x4_f32(false, a0, false, b1, (short)0, acc[0][1], false, false);
            acc[1][0] = __builtin_amdgcn_wmma_f32_16x16x4_f32(false, a1, false, b0, (short)0, acc[1][0], false, false);
            acc[1][1] = __builtin_amdgcn_wmma_f32_16x16x4_f32(false, a1, false, b1, (short)0, acc[1][1], false, false);
        }

        if (more) {
            float4 bv;
            bv.x = wv.x * (float)mv.x;
            bv.y = wv.y * (float)mv.y;
            bv.z = wv.z * (float)mv.z;
            bv.w = wv.w * (float)mv.w;
            *reinterpret_cast<float4*>(&Bsm[nxt][st_row * LDB + st_k]) = bv;
        }
        wait_asynccnt0();       // next-tile async A copies landed in LDS
        __syncthreads();        // + compiler-inserted dscnt wait for B stores
    }

    // ---- fused Pinsky-Rinzel RHS straight from the WMMA accumulators ----
    // C/D layout: VGPR r -> M = 8*hi + r, N = lane%16 (within each 16x16 subtile)
    #pragma unroll
    for (int si = 0; si < 2; ++si) {
        #pragma unroll
        for (int sj = 0; sj < 2; ++sj) {
            const int nn   = nbase + bn0 + sj * 16 + lm;       // neuron index
            const int msub = mbase + am0 + si * 16 + hi * 8;   // sample base
            #pragma unroll
            for (int r = 0; r < 8; ++r) {
                const int b = msub + r;
                const float syn = acc[si][sj][r];
                const size_t bn_idx = (size_t)b * NEUR + nn;
                const float ic = I_ext[bn_idx] * 10.0f + syn;   // I_ext/P + syn, P=0.1

                const size_t off = bn_idx * 8;
                const float4 y0 = *reinterpret_cast<const float4*>(y + off);
                const float4 y1 = *reinterpret_cast<const float4*>(y + off + 4);
                const float Vs = y0.x, Vd = y0.y, n_ = y0.z, h = y0.w;
                const float s  = y1.x, c  = y1.y, q  = y1.z, Ca = y1.w;

                // m_inf
                const float V1 = Vs + 46.9f;
                const float am = -0.32f * V1 / (__expf(-V1 * 0.25f) - 1.0f);
                const float V2 = Vs + 19.9f;
                const float bm = 0.28f * V2 / (__expf(V2 * 0.2f) - 1.0f);
                const float minf = am / (am + bm);
                // currents
                const float I_leak_s = 0.0004f * (Vs + 62.0f);
                const float I_Na     = 0.27f * minf * minf * h * (Vs - 60.0f);
                const float I_DR     = 0.24f * n_ * (Vs + 75.0f);
                const float I_ds     = 0.4f * (Vd - Vs);
                const float I_leak_d = 1e-5f * (Vd + 62.0f);
                const float I_Ca     = 0.008f * s * s * (Vd - 80.0f);
                const float I_AHP    = 0.005f * q * (Vd + 75.0f);
                const float chi      = fminf(Ca * (1.0f / 250.0f), 1.0f);
                const float I_C      = 0.009f * c * chi * (Vd + 75.0f);
                // voltage derivatives (1/C_m = 0.5, 1/P = 10, 1/(1-P) = 1/0.9)
                const float dVs = 0.5f * (-I_leak_s - I_Na - I_DR + I_ds * 10.0f + ic);
                const float dVd = 0.5f * (-I_leak_d - I_Ca - I_AHP - I_C - I_ds * (1.0f / 0.9f));
                // gates
                const float ah = 0.128f * __expf((-43.0f - Vs) * (1.0f / 18.0f));
                const float bh = 4.0f / (1.0f + __expf(-(Vs + 20.0f) * 0.2f));
                const float dh = ah * (1.0f - h) - bh * h;
                const float V3 = Vs + 24.9f;
                const float an = -0.016f * V3 / (__expf(-V3 * 0.2f) - 1.0f);
                const float bn_ = 0.25f * __expf(-(Vs + 40.0f) * 0.025f);
                const float dn = an * (1.0f - n_) - bn_ * n_;
                const float as_ = 1.6f / (1.0f + __expf(-0.072f * (Vd - 5.0f)));
                const float V6 = Vd + 8.9f;
                const float bs = 0.02f * V6 / (__expf(V6 * 0.2f) - 1.0f);
                const float dsv = as_ * (1.0f - s) - bs * s;
                // calcium-gated c channel (piecewise)
                const float V7 = Vd + 53.5f;
                const float V8 = Vd + 50.0f;
                const float e27 = __expf(-V7 * (1.0f / 27.0f));
                const bool lo = (Vd <= -10.0f);
                const float ac = lo ? 0.0527f * __expf(V8 * (1.0f / 11.0f) - V7 * (1.0f / 27.0f))
                                    : 2.0f * e27;
                const float bc = lo ? 2.0f * e27 - ac : 0.0f;
                const float dcv = ac * (1.0f - c) - bc * c;
                // q gate, calcium
                const float aq = fminf(2e-5f * Ca, 0.01f);
                const float dq = aq * (1.0f - q) - 0.001f * q;
                const float dCa = -0.004f * I_Ca - Ca * 0.125f;

                const float4 o0 = {dVs, dVd, dn, dh};
                const float4 o1 = {dsv, dcv, dq, dCa};
                *reinterpret_cast<float4*>(out + off)     = o0;
                *reinterpret_cast<float4*>(out + off + 4) = o1;
            }
        }
    }
}

extern "C" void kernel_launch(void* const* d_in, const int* in_sizes, int n_in,
                              void* d_out, int out_size, void* d_ws, size_t ws_size,
                              hipStream_t stream) {
    const float* y     = (const float*)d_in[0];   // [S, N, 8]
    const float* I_ext = (const float*)d_in[1];   // [S, N]
    const float* synI  = (const float*)d_in[2];   // [S, N]
    const float* w     = (const float*)d_in[3];   // [N, N]
    const int*   net   = (const int*)d_in[4];     // [N, N]
    float* out = (float*)d_out;                   // [S, N, 8]

    const int S = in_sizes[1] / NEUR;             // 8192
    dim3 grid(S / BM, NEUR / BN);                 // 64 x 16 blocks
    pr_fused_wmma_kernel<<<grid, 256, 0, stream>>>(y, I_ext, synI, w, net, out);
}